// RBFSubatom_36429912604709
// MI455X (gfx1250) — compile-verified
//
#include <hip/hip_runtime.h>
#include <hip/hip_bf16.h>
#include <math.h>

// Shapes: B=8, N=256, D=128, H=256, K=128
#define BB 8
#define NN 256
#define DD 128
#define HH 256
#define KK 128

typedef __attribute__((ext_vector_type(2))) float v2f;
typedef __attribute__((ext_vector_type(4))) float v4f;
typedef __attribute__((ext_vector_type(8))) float v8f;

// ---------------------------------------------------------------------------
// Kernel 1: hi = AE @ W1[:D] + b1 ; hj = AE @ W1[D:]
// AE: (B*N, D) = (2048, 128), W1: (2D, H) = (256, 256)
// One wave computes a 16x16 tile of both hi and hj (shared A fragment),
// using V_WMMA_F32_16X16X4_F32 over K=128 (32 steps, 2 wmma each).
// ---------------------------------------------------------------------------
__global__ void __launch_bounds__(256)
hij_wmma_kernel(const float* __restrict__ AE,
                const float* __restrict__ W1,
                const float* __restrict__ b1,
                float* __restrict__ hi,
                float* __restrict__ hj)
{
    const int lane = threadIdx.x & 31;
    const int wave = threadIdx.x >> 5;
    const int tile = blockIdx.x * 8 + wave;     // 0..2047
    const int tm   = tile >> 4;                 // row tile: 0..127
    const int tn   = tile & 15;                 // col tile: 0..15
    const int m0   = tm * 16;
    const int n0   = tn * 16;

    const int half = lane >> 4;                 // 0: K0/K1, 1: K2/K3
    const int l16  = lane & 15;

    // A fragment source: row (m0+l16), k offset 2*half (+0,+1)
    const float* Arow = AE + (size_t)(m0 + l16) * DD + 2 * half;
    // B fragments: W1a = W1[0:128], W1b = W1[128:256]; column n0+l16
    const float* Ba = W1 + (size_t)0   * HH + n0 + l16;
    const float* Bb = W1 + (size_t)DD  * HH + n0 + l16;

    v8f acc_i = {};
    v8f acc_j = {};

#pragma unroll
    for (int k = 0; k < DD; k += 4) {
        // A 16x4 f32: lane<16 -> (K=k,k+1), lane>=16 -> (K=k+2,k+3)
        v2f a = *(const v2f*)(Arow + k);        // 8B aligned (even float index)
        const int kk = k + 2 * half;
        v2f bA, bB;
        bA.x = Ba[(size_t)(kk + 0) * HH];
        bA.y = Ba[(size_t)(kk + 1) * HH];
        bB.x = Bb[(size_t)(kk + 0) * HH];
        bB.y = Bb[(size_t)(kk + 1) * HH];
        acc_i = __builtin_amdgcn_wmma_f32_16x16x4_f32(
            false, a, false, bA, (short)0, acc_i, false, false);
        acc_j = __builtin_amdgcn_wmma_f32_16x16x4_f32(
            false, a, false, bB, (short)0, acc_j, false, false);
    }

    // Fold b1 into hi so the edge kernel does h = relu(hi' + hj).
    const float bias1 = b1[n0 + l16];

    // D layout: lane holds D[r + 8*(lane/16)][lane%16] in VGPR r.
#pragma unroll
    for (int r = 0; r < 8; ++r) {
        const int m = m0 + r + 8 * half;
        hi[(size_t)m * HH + n0 + l16] = acc_i[r] + bias1;
        hj[(size_t)m * HH + n0 + l16] = acc_j[r];
    }
}

// ---------------------------------------------------------------------------
// Kernel 2: fused edge-MLP + RBF expansion.
// One block = one (b, 16x16 i,j) tile. 256 threads.
//   Phase 1: thread-per-edge: v = ((relu(hi+hj)·W2 + b2) * scale) applied to x
//   Phase 2: wave-per-edge coalesced 128b NT stores of exp(-(v-mean)^2*|temp|)
// Output (268 MB) is write-once and exceeds the 192 MB L2 -> stream it with
// non-temporal stores so L2 stays filled with the re-used hi/hj tiles.
// ---------------------------------------------------------------------------
#define ROWP 260   // padded LDS row stride (floats): 260 % 64 = 4 -> no bank conflicts

__global__ void __launch_bounds__(256)
edge_rbf_kernel(const float* __restrict__ x,
                const float* __restrict__ hi,
                const float* __restrict__ hj,
                const float* __restrict__ W2,
                const float* __restrict__ b2,
                const float* __restrict__ means,
                const float* __restrict__ temps,
                float* __restrict__ out)
{
    __shared__ float s_hi[16 * ROWP];
    __shared__ float s_hj[16 * ROWP];
    __shared__ float s_w2[2 * HH];
    __shared__ float s_v[256];

    const int t     = threadIdx.x;
    const int b     = blockIdx.y;
    const int itile = blockIdx.x >> 4;
    const int jtile = blockIdx.x & 15;
    const int i0    = itile * 16;
    const int j0    = jtile * 16;

    // ---- cooperative loads into LDS (padded rows) ----
    {
        const int row = t >> 4;        // 0..15
        const int col = t & 15;        // 16 threads per row, 4 float4 each
        const v4f* srcI = (const v4f*)(hi + (size_t)(b * NN + i0 + row) * HH);
        const v4f* srcJ = (const v4f*)(hj + (size_t)(b * NN + j0 + row) * HH);
        v4f* dI = (v4f*)(s_hi + row * ROWP);
        v4f* dJ = (v4f*)(s_hj + row * ROWP);
#pragma unroll
        for (int rep = 0; rep < 4; ++rep) {
            dI[col + 16 * rep] = srcI[col + 16 * rep];
            dJ[col + 16 * rep] = srcJ[col + 16 * rep];
        }
        if (t < 128) ((v4f*)s_w2)[t] = ((const v4f*)W2)[t];
    }
    __syncthreads();

    // ---- phase 1: one edge per thread ----
    {
        const int ti = t >> 4;
        const int tj = t & 15;
        const v4f* a4 = (const v4f*)(s_hi + ti * ROWP);
        const v4f* c4 = (const v4f*)(s_hj + tj * ROWP);
        const v4f* w4 = (const v4f*)s_w2;

        float acc0 = 0.f, acc1 = 0.f;
#pragma unroll 8
        for (int h4 = 0; h4 < HH / 4; ++h4) {
            const v4f a   = a4[h4];
            const v4f c   = c4[h4];
            const v4f w01 = w4[2 * h4 + 0];      // (h0.o0,h0.o1,h1.o0,h1.o1)
            const v4f w23 = w4[2 * h4 + 1];
            const float h0 = fmaxf(a.x + c.x, 0.f);
            const float h1 = fmaxf(a.y + c.y, 0.f);
            const float h2 = fmaxf(a.z + c.z, 0.f);
            const float h3 = fmaxf(a.w + c.w, 0.f);
            acc0 = fmaf(h0, w01.x, acc0); acc1 = fmaf(h0, w01.y, acc1);
            acc0 = fmaf(h1, w01.z, acc0); acc1 = fmaf(h1, w01.w, acc1);
            acc0 = fmaf(h2, w23.x, acc0); acc1 = fmaf(h2, w23.y, acc1);
            acc0 = fmaf(h3, w23.z, acc0); acc1 = fmaf(h3, w23.w, acc1);
        }
        const float scale = 0.0625f;             // 1/sqrt(2*D) = 1/16
        const float mul   = (acc0 + b2[0]) * scale;
        const float bia   = (acc1 + b2[1]) * scale;
        // x is read exactly once per element: non-temporal load
        const float xv = __builtin_nontemporal_load(
            x + (size_t)(b * NN + i0 + ti) * NN + (j0 + tj));
        s_v[t] = fmaf(mul, xv, bia);
    }
    __syncthreads();

    // ---- phase 2: wave-per-edge coalesced RBF expansion ----
    {
        const int k4 = (t & 31) * 4;             // 4 consecutive k per thread
        const int eg = t >> 5;                   // edge group 0..7 (one per wave)
        v4f m4 = *(const v4f*)(means + k4);
        v4f t4 = *(const v4f*)(temps + k4);
        t4.x = fabsf(t4.x); t4.y = fabsf(t4.y);
        t4.z = fabsf(t4.z); t4.w = fabsf(t4.w);

#pragma unroll 4
        for (int it = 0; it < 32; ++it) {
            const int edge = it * 8 + eg;        // 0..255
            const float v  = s_v[edge];          // broadcast within a wave
            v4f r;
            float d;
            d = v - m4.x; r.x = __expf(-d * d * t4.x);
            d = v - m4.y; r.y = __expf(-d * d * t4.y);
            d = v - m4.z; r.z = __expf(-d * d * t4.z);
            d = v - m4.w; r.w = __expf(-d * d * t4.w);
            const int ei = edge >> 4;
            const int ej = edge & 15;
            v4f* dst = (v4f*)(out +
                ((size_t)(b * NN + i0 + ei) * NN + (j0 + ej)) * (size_t)KK + k4);
            __builtin_nontemporal_store(r, dst); // 512B contiguous per wave, TH=NT
        }
    }
}

// ---------------------------------------------------------------------------
extern "C" void kernel_launch(void* const* d_in, const int* in_sizes, int n_in,
                              void* d_out, int out_size, void* d_ws, size_t ws_size,
                              hipStream_t stream) {
    const float* x     = (const float*)d_in[0];   // (B,N,N)
    const float* ae    = (const float*)d_in[1];   // (B,N,D)
    const float* W1    = (const float*)d_in[2];   // (2D,H)
    const float* b1    = (const float*)d_in[3];   // (H)
    const float* W2    = (const float*)d_in[4];   // (H,2)
    const float* b2    = (const float*)d_in[5];   // (2)
    const float* means = (const float*)d_in[6];   // (K)
    const float* temps = (const float*)d_in[7];   // (K)
    float* out = (float*)d_out;

    float* hi = (float*)d_ws;                     // (2048,256) = 2 MB
    float* hj = hi + (size_t)BB * NN * HH;        // (2048,256) = 2 MB

    // 2048 wave-tiles / 8 waves per block -> 256 blocks
    hij_wmma_kernel<<<256, 256, 0, stream>>>(ae, W1, b1, hi, hj);

    // 16x16 j-tiles * 16 i-tiles = 256 tiles per batch, 8 batches
    edge_rbf_kernel<<<dim3(256, BB), 256, 0, stream>>>(
        x, hi, hj, W2, b2, means, temps, out);
}